// Attention_16552803959178
// MI455X (gfx1250) — compile-verified
//
#include <hip/hip_runtime.h>
#include <hip/hip_bf16.h>
#include <math.h>

#define N_NODES 100000
#define N_EDGES 3200000
#define HDIM    128
#define EDIM    16

typedef __attribute__((ext_vector_type(2))) float v2f;
typedef __attribute__((ext_vector_type(8))) float v8f;

// order-preserving float<->uint encoding so atomicMax(uint) == float max
__device__ __forceinline__ unsigned f2ord(float f) {
    unsigned u = __float_as_uint(f);
    return (u & 0x80000000u) ? ~u : (u | 0x80000000u);
}
__device__ __forceinline__ float ord2f(unsigned u) {
    return (u & 0x80000000u) ? __uint_as_float(u & 0x7fffffffu)
                             : __uint_as_float(~u);
}

__global__ void init_kernel(float* __restrict__ out, unsigned* __restrict__ m0,
                            unsigned* __restrict__ m1, float* __restrict__ d0,
                            float* __restrict__ d1) {
    const long long total = (long long)N_NODES * HDIM;
    for (long long i = blockIdx.x * 256ll + threadIdx.x; i < total;
         i += (long long)gridDim.x * 256ll) {
        out[i] = 0.0f;
        if (i < N_NODES) {
            m0[i] = 0u;   // encodes "below any finite float"
            m1[i] = 0u;
            d0[i] = 0.0f;
            d1[i] = 0.0f;
        }
    }
}

// One wave computes s0 for 16 edges: diag(Qsel * Ksel^T) via chained
// V_WMMA_F32_16X16X4_F32 over K=128, plus the 16-dim eig dot the same way.
__global__ void __launch_bounds__(256)
score_kernel(const float* __restrict__ q, const float* __restrict__ k,
             const float* __restrict__ eigs, const float* __restrict__ lambda0,
             const float* __restrict__ pw, const int* __restrict__ indices,
             const int* __restrict__ ptype, float* __restrict__ s0out,
             unsigned* __restrict__ m0, unsigned* __restrict__ m1) {
    const int lane = threadIdx.x & 31;
    const int wave = blockIdx.x * (blockDim.x >> 5) + (threadIdx.x >> 5);
    const long long ebase = (long long)wave * 16;
    if (ebase >= N_EDGES) return;            // wave-uniform: EXEC stays full

    const int e      = (int)ebase + (lane & 15);
    const bool valid = (e < N_EDGES);
    const int ec     = valid ? e : (N_EDGES - 1);
    const int row    = indices[ec];              // destination (segment id)
    const int col    = indices[N_EDGES + ec];    // source
    const int half   = lane >> 4;                // 0: K even pair, 1: K odd pair

    // ---- q . k over K=128 : 32 x wmma_f32_16x16x4_f32, dots on the diagonal
    // A 16x4 f32 layout: lanes0-15 M=0-15 {K0,K1}; lanes16-31 {K2,K3}
    // B 4x16  f32 layout: row-striped, symmetric pair per lane half
    v8f c = {};
    const float* qp = q + (long long)row * HDIM + half * 2;
    const float* kp = k + (long long)col * HDIM + half * 2;
#pragma unroll
    for (int kc = 0; kc < HDIM / 4; ++kc) {
        v2f a = *(const v2f*)(qp + kc * 4);
        v2f b = *(const v2f*)(kp + kc * 4);
        c = __builtin_amdgcn_wmma_f32_16x16x4_f32(false, a, false, b,
                                                  (short)0, c, false, false);
    }

    // ---- eigs . eigs over K=16 : 4 wmma ops
    v8f ce = {};
    const float* ep0 = eigs + (long long)row * EDIM + half * 2;
    const float* ep1 = eigs + (long long)col * EDIM + half * 2;
#pragma unroll
    for (int kc = 0; kc < EDIM / 4; ++kc) {
        v2f a = *(const v2f*)(ep0 + kc * 4);
        v2f b = *(const v2f*)(ep1 + kc * 4);
        ce = __builtin_amdgcn_wmma_f32_16x16x4_f32(false, a, false, b,
                                                   (short)0, ce, false, false);
    }

    // ---- extract diagonal: elem (m,m) lives in VGPR m&7, lane m (m<8) / m+16
    float x = 0.0f, y = 0.0f;
#pragma unroll
    for (int m = 0; m < 16; ++m) {
        const int src = (m < 8) ? m : (m + 16);
        float dx = __shfl(c[m & 7], src, 32);
        float dy = __shfl(ce[m & 7], src, 32);
        if (lane == m) { x = dx; y = dy; }
    }

    const float lam = __expf(lambda0[0]);
    const float s0  = x * 0.088388347648318447f /* 1/sqrt(128) */ + lam * y;

    if (lane < 16 && valid) {
        s0out[e] = s0;
        atomicMax(&m0[row], f2ord(s0));
        const float s1 = pw[ptype[e]];
        atomicMax(&m1[row], f2ord(s1));
    }
}

__global__ void denom_kernel(const float* __restrict__ s0,
                             const int* __restrict__ indices,
                             const int* __restrict__ ptype,
                             const float* __restrict__ pw,
                             const unsigned* __restrict__ m0u,
                             const unsigned* __restrict__ m1u,
                             float* __restrict__ d0, float* __restrict__ d1) {
    for (long long e = blockIdx.x * 256ll + threadIdx.x; e < N_EDGES;
         e += (long long)gridDim.x * 256ll) {
        const int r = indices[e];
        unsafeAtomicAdd(&d0[r], __expf(s0[e] - ord2f(m0u[r])));
        unsafeAtomicAdd(&d1[r], __expf(pw[ptype[e]] - ord2f(m1u[r])));
    }
}

// 32 lanes per edge; each lane owns a float4 slice of the H=128 value row.
__global__ void scatter_kernel(const float* __restrict__ v,
                               const float* __restrict__ s0,
                               const int* __restrict__ indices,
                               const int* __restrict__ ptype,
                               const float* __restrict__ pw,
                               const unsigned* __restrict__ m0u,
                               const unsigned* __restrict__ m1u,
                               const float* __restrict__ d0,
                               const float* __restrict__ d1,
                               float* __restrict__ out) {
    const long long total = (long long)N_EDGES * 32;
    for (long long i = blockIdx.x * 256ll + threadIdx.x; i < total;
         i += (long long)gridDim.x * 256ll) {
        const int e    = (int)(i >> 5);
        const int lane = (int)(i & 31);
        const int r    = indices[e];
        const int cc   = indices[N_EDGES + e];
        const float a0 = __expf(s0[e] - ord2f(m0u[r])) / d0[r];
        const float a1 = __expf(pw[ptype[e]] - ord2f(m1u[r])) / d1[r];
        const float w  = 0.5f * (a0 + a1);
        const float4 vv = *(const float4*)(v + (long long)cc * HDIM + lane * 4);
        float* op = out + (long long)r * HDIM + lane * 4;
        unsafeAtomicAdd(op + 0, w * vv.x);
        unsafeAtomicAdd(op + 1, w * vv.y);
        unsafeAtomicAdd(op + 2, w * vv.z);
        unsafeAtomicAdd(op + 3, w * vv.w);
    }
}

extern "C" void kernel_launch(void* const* d_in, const int* in_sizes, int n_in,
                              void* d_out, int out_size, void* d_ws, size_t ws_size,
                              hipStream_t stream) {
    const float* q       = (const float*)d_in[0];
    const float* k       = (const float*)d_in[1];
    const float* v       = (const float*)d_in[2];
    const float* eigs    = (const float*)d_in[3];
    const float* lambda0 = (const float*)d_in[4];
    const float* pw      = (const float*)d_in[5];
    const int*   indices = (const int*)d_in[6];
    const int*   ptype   = (const int*)d_in[7];
    float* out = (float*)d_out;

    // workspace layout: s0[E] | m0[N] | m1[N] | d0[N] | d1[N]  (~14.4 MB)
    float*    s0 = (float*)d_ws;
    unsigned* m0 = (unsigned*)(s0 + N_EDGES);
    unsigned* m1 = m0 + N_NODES;
    float*    d0 = (float*)(m1 + N_NODES);
    float*    d1 = d0 + N_NODES;

    init_kernel<<<50000, 256, 0, stream>>>(out, m0, m1, d0, d1);

    // E/16 waves, 8 waves (256 threads) per block
    const int score_blocks = (N_EDGES / 16 + 7) / 8;
    score_kernel<<<score_blocks, 256, 0, stream>>>(q, k, eigs, lambda0, pw,
                                                   indices, ptype, s0, m0, m1);

    denom_kernel<<<8192, 256, 0, stream>>>(s0, indices, ptype, pw, m0, m1, d0, d1);

    scatter_kernel<<<32768, 256, 0, stream>>>(v, s0, indices, ptype, pw,
                                              m0, m1, d0, d1, out);
}